// MyLoss_29420525977942
// MI455X (gfx1250) — compile-verified
//
#include <hip/hip_runtime.h>
#include <hip/hip_bf16.h>
#include <stdint.h>

// ---------------------------------------------------------------------------
// Masked per-class MSE loss (10 classes) for MI455X (gfx1250).
//
// Bandwidth-bound: 96 MiB in / 23.3 TB/s ~= 4.3 us floor. Strategy:
//   * Stream tiles global->LDS with CDNA5 async copies (ASYNCcnt), 3 deep.
//   * Per-thread-column LDS histogram (hist[class][tid]) -> ds_add_f32 with
//     zero bank conflicts, 2 LDS atomics per pixel.
//   * Block reduction: 6-step tree (256 -> 4 columns), then the final 4->1
//     horizontal sum via V_WMMA_F32_16X16X4_F32 (A = ones, B[k][class]) on
//     wave 0 -- matrix-unit cross-lane reduction, exact f32, fixed order.
//   * Deterministic two-stage reduction (no global float atomics).
// ---------------------------------------------------------------------------

#define NTHREADS 256
#define TILE     1024          // pixels per tile (4 KB per array)
#define NCLS     10
#define HROWS    11            // 10 classes + 1 dump row for masked-out pixels

typedef __attribute__((ext_vector_type(2))) float v2f;
typedef __attribute__((ext_vector_type(8))) float v8f;

// One async b128 copy: each lane moves 16 bytes global -> LDS. Per wave this
// is a single instruction tracked by ASYNCcnt.
__device__ __forceinline__ void async_b128(uint32_t lds_byte_off, const void* gptr) {
    asm volatile("global_load_async_to_lds_b128 %0, %1, off"
                 :: "v"(lds_byte_off), "v"((uint64_t)(uintptr_t)gptr)
                 : "memory");
}

// Issue one tile (1024 pixels) of all three arrays: 3 ASYNCcnt ticks per wave.
__device__ __forceinline__ void issue_tile(const float* outs, const int* tgts,
                                           const int* msks, long long pix0,
                                           uint32_t lo, uint32_t lt, uint32_t lm) {
    async_b128(lo, (const void*)(outs + pix0));
    async_b128(lt, (const void*)(tgts + pix0));
    async_b128(lm, (const void*)(msks + pix0));
}

__device__ __forceinline__ void accum_pixel(float o, int t, int m, int tid,
                                            float* s_sum, float* s_cnt) {
    float d = o - (float)t;
    // Fold mask into the class id: invalid/masked-out pixels land in dump row.
    int c = (m == 1 && (unsigned)t < (unsigned)NCLS) ? t : NCLS;
    atomicAdd(&s_sum[c * NTHREADS + tid], d * d);   // ds_add_f32, private slot
    atomicAdd(&s_cnt[c * NTHREADS + tid], 1.0f);    // ds_add_f32, private slot
}

__global__ __launch_bounds__(NTHREADS)
void loss_main(const float* __restrict__ outs, const int* __restrict__ tgts,
               const int* __restrict__ msks, float* __restrict__ ws,
               int tiles_total, int nwg) {
    __shared__ float s_out[3][TILE];
    __shared__ int   s_tgt[3][TILE];
    __shared__ int   s_msk[3][TILE];
    __shared__ float s_sum[HROWS * NTHREADS];
    __shared__ float s_cnt[HROWS * NTHREADS];

    const int tid = threadIdx.x;
    const int wg  = blockIdx.x;

    for (int i = tid; i < HROWS * NTHREADS; i += NTHREADS) {
        s_sum[i] = 0.0f;
        s_cnt[i] = 0.0f;
    }

    // LDS byte addresses for this thread's 16-byte staging slot in buffer 0.
    const uint32_t lo0 = (uint32_t)(uintptr_t)&s_out[0][0] + (uint32_t)tid * 16u;
    const uint32_t lt0 = (uint32_t)(uintptr_t)&s_tgt[0][0] + (uint32_t)tid * 16u;
    const uint32_t lm0 = (uint32_t)(uintptr_t)&s_msk[0][0] + (uint32_t)tid * 16u;
    const uint32_t bstr = TILE * 4u;   // bytes per buffer

    // Tiles for this workgroup: wg, wg+nwg, wg+2*nwg, ...
    const int K = (wg < tiles_total) ? ((tiles_total - 1 - wg) / nwg + 1) : 0;

    // Prologue: fill the 3-deep pipeline.
    for (int k = 0; k < 3 && k < K; ++k) {
        uint32_t off = (uint32_t)k * bstr;
        issue_tile(outs, tgts, msks,
                   ((long long)wg + (long long)k * nwg) * TILE + (long long)tid * 4,
                   lo0 + off, lt0 + off, lm0 + off);
    }

    for (int k = 0; k < K; ++k) {
        // Tiles issued so far reach min(k+2, K-1); wait until only the loads
        // issued AFTER tile k remain outstanding (3 per tile, in-order).
        int beyond = ((k + 2 < K) ? (k + 2) : (K - 1)) - k;
        if (beyond >= 2)      asm volatile("s_wait_asynccnt 6" ::: "memory");
        else if (beyond == 1) asm volatile("s_wait_asynccnt 3" ::: "memory");
        else                  asm volatile("s_wait_asynccnt 0" ::: "memory");
        __syncthreads();  // all waves' tile-k data now in LDS

        const int buf = k % 3;
        const float4 o4 = *reinterpret_cast<const float4*>(&s_out[buf][tid * 4]);
        const int4   t4 = *reinterpret_cast<const int4*>(&s_tgt[buf][tid * 4]);
        const int4   m4 = *reinterpret_cast<const int4*>(&s_msk[buf][tid * 4]);

        accum_pixel(o4.x, t4.x, m4.x, tid, s_sum, s_cnt);
        accum_pixel(o4.y, t4.y, m4.y, tid, s_sum, s_cnt);
        accum_pixel(o4.z, t4.z, m4.z, tid, s_sum, s_cnt);
        accum_pixel(o4.w, t4.w, m4.w, tid, s_sum, s_cnt);

        __syncthreads();  // everyone done reading buf before we overwrite it

        if (k + 3 < K) {  // refill this buffer with tile k+3 ( (k+3)%3 == buf )
            uint32_t off = (uint32_t)buf * bstr;
            issue_tile(outs, tgts, msks,
                       ((long long)wg + (long long)(k + 3) * nwg) * TILE + (long long)tid * 4,
                       lo0 + off, lt0 + off, lm0 + off);
        }
    }

    // Fixed-order tree reduction over thread columns: 256 -> 4 columns.
    __syncthreads();
    for (int s = NTHREADS / 2; s >= 4; s >>= 1) {
        if (tid < s) {
#pragma unroll
            for (int c = 0; c < NCLS; ++c) {
                s_sum[c * NTHREADS + tid] += s_sum[c * NTHREADS + tid + s];
                s_cnt[c * NTHREADS + tid] += s_cnt[c * NTHREADS + tid + s];
            }
        }
        __syncthreads();
    }

    // Final 4 -> 1 reduction on the matrix unit (wave 0, EXEC all-ones):
    //   D = ones(16x4) x B(4x16) + 0, with B[k][c] = partial k of class c,
    // so D[0][c] = total of class c. One WMMA for sums, one for counts.
    // B f32 layout (ISA 7.12.2): VGPR0 = rows K0 (lanes 0-15) / K1 (16-31),
    //                            VGPR1 = rows K2 (lanes 0-15) / K3 (16-31).
    if ((tid >> 5) == 0) {              // wave 0 only (uniform branch)
        const int half = tid >> 4;       // 0: rows K0/K2, 1: rows K1/K3
        const int col  = tid & 15;       // column = class id
        const bool valid = (col < NCLS);

        v2f b_sum, b_cnt;
        b_sum[0] = valid ? s_sum[col * NTHREADS + half]     : 0.0f;  // K = half
        b_sum[1] = valid ? s_sum[col * NTHREADS + 2 + half] : 0.0f;  // K = 2+half
        b_cnt[0] = valid ? s_cnt[col * NTHREADS + half]     : 0.0f;
        b_cnt[1] = valid ? s_cnt[col * NTHREADS + 2 + half] : 0.0f;

        v2f a_ones;
        a_ones[0] = 1.0f;
        a_ones[1] = 1.0f;
        v8f c0 = {};

        v8f d_sum = __builtin_amdgcn_wmma_f32_16x16x4_f32(
            false, a_ones, false, b_sum, (short)0, c0, false, false);
        v8f d_cnt = __builtin_amdgcn_wmma_f32_16x16x4_f32(
            false, a_ones, false, b_cnt, (short)0, c0, false, false);

        // D row M=0 lives in VGPR0 of lanes 0-15: lane c holds class-c total.
        if (tid < NCLS) {
            ws[wg * 20 + tid]      = d_sum[0];
            ws[wg * 20 + 10 + tid] = d_cnt[0];
        }
    }
}

__global__ __launch_bounds__(NTHREADS)
void loss_finalize(const float* __restrict__ ws, int nwg,
                   const float* __restrict__ outs, const int* __restrict__ tgts,
                   const int* __restrict__ msks, long long tail_start,
                   long long tail_n, float* __restrict__ out) {
    __shared__ float s_sum[HROWS * NTHREADS];
    __shared__ float s_cnt[HROWS * NTHREADS];
    __shared__ float s_l4e[NCLS];
    const int tid = threadIdx.x;

    for (int i = tid; i < HROWS * NTHREADS; i += NTHREADS) {
        s_sum[i] = 0.0f;
        s_cnt[i] = 0.0f;
    }
    __syncthreads();

    // Tail pixels not covered by full tiles (usually 0; also the fallback
    // path if the workspace is too small for any main-kernel blocks).
    for (long long i = tid; i < tail_n; i += NTHREADS) {
        float o = outs[tail_start + i];
        int   t = tgts[tail_start + i];
        int   m = msks[tail_start + i];
        float d = o - (float)t;
        int c = (m == 1 && (unsigned)t < (unsigned)NCLS) ? t : NCLS;
        atomicAdd(&s_sum[c * NTHREADS + tid], d * d);
        atomicAdd(&s_cnt[c * NTHREADS + tid], 1.0f);
    }
    __syncthreads();

    for (int s = NTHREADS / 2; s >= 1; s >>= 1) {
        if (tid < s) {
#pragma unroll
            for (int c = 0; c < NCLS; ++c) {
                s_sum[c * NTHREADS + tid] += s_sum[c * NTHREADS + tid + s];
                s_cnt[c * NTHREADS + tid] += s_cnt[c * NTHREADS + tid + s];
            }
        }
        __syncthreads();
    }

    if (tid < NCLS) {
        float s = s_sum[tid * NTHREADS];
        float n = s_cnt[tid * NTHREADS];
        for (int w = 0; w < nwg; ++w) {       // fixed order -> deterministic
            s += ws[w * 20 + tid];
            n += ws[w * 20 + 10 + tid];
        }
        float l = (n > 0.0f) ? s / fmaxf(n, 1.0f) : 0.0f;
        s_l4e[tid]    = l;
        out[1 + tid]  = l;   // loss4each
        out[11 + tid] = n;   // class_n
    }
    __syncthreads();
    if (tid == 0) {
        float acc = 0.0f;
        for (int c = 0; c < NCLS; ++c) acc += 0.1f * s_l4e[c];
        out[0] = acc;        // loss
    }
}

extern "C" void kernel_launch(void* const* d_in, const int* in_sizes, int n_in,
                              void* d_out, int out_size, void* d_ws, size_t ws_size,
                              hipStream_t stream) {
    const float* outs = (const float*)d_in[0];
    const int*   tgts = (const int*)d_in[1];
    const int*   msks = (const int*)d_in[2];
    float*       out  = (float*)d_out;
    float*       ws   = (float*)d_ws;

    const long long N = (long long)in_sizes[0];   // 32*512*512 = 8,388,608
    int tiles_total = (int)(N / TILE);

    int nwg = tiles_total < 1024 ? tiles_total : 1024;
    int max_by_ws = (int)(ws_size / (20 * sizeof(float)));
    if (nwg > max_by_ws) nwg = max_by_ws;

    long long tail_start = (long long)tiles_total * TILE;
    long long tail_n     = N - tail_start;

    if (nwg > 0) {
        loss_main<<<nwg, NTHREADS, 0, stream>>>(outs, tgts, msks, ws,
                                                tiles_total, nwg);
    } else {
        // No workspace for partials: finalize does everything (slow, correct).
        tail_start = 0;
        tail_n     = N;
    }
    loss_finalize<<<1, NTHREADS, 0, stream>>>(ws, nwg, outs, tgts, msks,
                                              tail_start, tail_n, out);
}